// EmmaAttention_15152644620653
// MI455X (gfx1250) — compile-verified
//
#include <hip/hip_runtime.h>
#include <math.h>

// Elementwise EMA merge (EmmaAttention forward), MI455X / gfx1250.
//
// Memory-bound: ~1.24 GB moved per call vs ~0.3 GFLOP -> HBM roofline
// (~53 us at 23.3 TB/s). Optimal path is 128-bit coalesced vector memory
// with non-temporal cache hints on the zero-reuse streams; WMMA/TDM have
// no role in an elementwise kernel.

typedef __attribute__((ext_vector_type(4))) float v4f;

__global__ __launch_bounds__(256) void emma_ema_merge_kernel(
    const v4f* __restrict__ x4,      // [N,H,D] as float4
    const float* __restrict__ max_a, // [N,H]
    const v4f* __restrict__ his_x4,  // [N,H,D] as float4
    const float* __restrict__ his_m, // [N,H]
    const float* __restrict__ agg_n, // [N]
    const float* __restrict__ inv_w, // [1]
    v4f* __restrict__ out4,          // [N,H,D] as float4
    long long total4)                // N*H*D/4
{
    long long v = (long long)blockIdx.x * blockDim.x + threadIdx.x;
    if (v >= total4) return;

    // D = 64 floats -> 16 float4 per (n,h) row; H = 8 rows per n.
    long long row = v >> 4;
    long long n   = row >> 3;

    // beta = clip(1 - inv_w * agg_n, 0, 1)
    float beta = 1.0f - inv_w[0] * agg_n[n];
    beta = fminf(fmaxf(beta, 0.0f), 1.0f);

    // log-sum-exp stabilized merge weights (shared by the 16 lanes of this
    // row; recomputed per lane -- VALU is free at 0.25 FLOP/byte).
    float hm = his_m[row];
    float ma = max_a[row];
    float mm = fmaxf(ma, hm);

    float pd = hm - ma;            // == hm - mm when ma >= hm, else 0
    pd = hm - mm;
    if (pd != pd) pd = -INFINITY;  // nan_to_num(nan = -inf)
    float qd = ma - mm;
    if (qd != qd) qd = -INFINITY;

    float p = __expf(pd) * beta;   // v_exp_f32 path; exp(-inf) -> 0
    float q = __expf(qd);
    float t = fmaxf(p + q, 1.0f);
    float rt = 1.0f / t;
    p *= rt;
    q *= rt;

    // Zero-reuse 1.24 GB streams: non-temporal 128-bit loads/stores so the
    // L2 keeps only the reused per-row scalar arrays.
    v4f hx = __builtin_nontemporal_load(&his_x4[v]);
    v4f xx = __builtin_nontemporal_load(&x4[v]);
    v4f o  = hx * p + xx * q;      // fused to v_fma by -O3 contract
    __builtin_nontemporal_store(o, &out4[v]);
}

extern "C" void kernel_launch(void* const* d_in, const int* in_sizes, int n_in,
                              void* d_out, int out_size, void* d_ws, size_t ws_size,
                              hipStream_t stream) {
    (void)n_in; (void)d_ws; (void)ws_size; (void)out_size;

    const v4f*   x4     = (const v4f*)  d_in[0];  // x      [N,H,D]
    const float* max_a  = (const float*)d_in[1];  // max_a  [N,H]
    const v4f*   his_x4 = (const v4f*)  d_in[2];  // his_x  [N,H,D]
    const float* his_m  = (const float*)d_in[3];  // his_m  [N,H]
    const float* agg_n  = (const float*)d_in[4];  // agg_n  [N]
    const float* inv_w  = (const float*)d_in[5];  // inv_w  [1]
    v4f*         out4   = (v4f*)d_out;

    // D = 64 guarantees divisibility by 4.
    long long total  = (long long)in_sizes[0];    // N*H*D elements
    long long total4 = total >> 2;

    const int block = 256;                        // 8 wave32s per workgroup
    long long grid  = (total4 + block - 1) / block;

    emma_ema_merge_kernel<<<(unsigned int)grid, block, 0, stream>>>(
        x4, max_a, his_x4, his_m, agg_n, inv_w, out4, total4);
}